// RegionalPRNNLayer_34651796144296
// MI455X (gfx1250) — compile-verified
//
#include <hip/hip_runtime.h>
#include <hip/hip_bf16.h>
#include <math.h>

typedef __attribute__((ext_vector_type(2))) float v2f;
typedef __attribute__((ext_vector_type(8))) float v8f;

// heaviside(x) = (tanh(5x)+1)/2.  gfx1250 has native v_tanh_f32.
__device__ __forceinline__ float heav(float x) {
#if __has_builtin(__builtin_amdgcn_tanhf)
    return __builtin_amdgcn_tanhf(5.0f * x) * 0.5f + 0.5f;
#else
    // == sigmoid(10x)
    return __builtin_amdgcn_rcpf(1.0f + __expf(-10.0f * x));
#endif
}

// ---------------------------------------------------------------------------
// Kernel A: per-basin parameter MLP via V_WMMA_F32_16X16X4_F32.
// One wave computes a 16-basin tile. A-tile and w1 are staged through
// zero-padded LDS (K: 27 -> 28) so fragment reads are unconditional
// (no exec-mask branch ladders).
// params layout (SoA): params[j*Btot + b], j = tmin,tmax,ddf,f,smax,qmax.
// ---------------------------------------------------------------------------
__global__ void __launch_bounds__(32)
mlp_params_kernel(const float* __restrict__ attrs,
                  const float* __restrict__ w1,
                  const float* __restrict__ b1,
                  const float* __restrict__ w2,
                  const float* __restrict__ b2,
                  float* __restrict__ params,
                  int Btot) {
    __shared__ float lds_a[16 * 28];   // A tile, K padded 27->28 with zeros
    __shared__ float lds_b[28 * 32];   // w1,    K padded 27->28 with zeros
    __shared__ float lds_h[16 * 32];   // hidden activations

    const int lane = threadIdx.x;      // 0..31, EXEC all-ones (WMMA requirement)
    const int half = lane >> 4;        // 0 or 1
    const int lm   = lane & 15;
    const int row0 = blockIdx.x * 16;  // 16 basins per tile

    // Zero-fill pads, then coalesced copy-in.
    for (int i = lane; i < 16 * 28; i += 32) lds_a[i] = 0.0f;
    for (int i = lane; i < 28 * 32; i += 32) lds_b[i] = 0.0f;
    __syncthreads();
    for (int i = lane; i < 16 * 27; i += 32) {
        const int r = i / 27, c = i - r * 27;
        lds_a[r * 28 + c] = attrs[row0 * 27 + i];      // contiguous global read
    }
    for (int i = lane; i < 27 * 32; i += 32)
        lds_b[i] = w1[i];                              // same row-major layout
    __syncthreads();

    v8f c0 = {};   // N = 0..15
    v8f c1 = {};   // N = 16..31
    const int koff = half * 2;  // 32-bit A/B layout: lanes16-31 hold K+2,K+3

    #pragma unroll
    for (int ks = 0; ks < 7; ++ks) {   // K = 28 = 7 x K4
        const int k0 = ks * 4 + koff;
        v2f a, bl, bh;
        a.x  = lds_a[lm * 28 + k0];
        a.y  = lds_a[lm * 28 + k0 + 1];
        bl.x = lds_b[(k0)     * 32 + lm];
        bl.y = lds_b[(k0 + 1) * 32 + lm];
        bh.x = lds_b[(k0)     * 32 + 16 + lm];
        bh.y = lds_b[(k0 + 1) * 32 + 16 + lm];
        c0 = __builtin_amdgcn_wmma_f32_16x16x4_f32(false, a, false, bl,
                                                   (short)0, c0, false, false);
        c1 = __builtin_amdgcn_wmma_f32_16x16x4_f32(false, a, false, bh,
                                                   (short)0, c1, false, false);
    }

    // C/D layout: VGPR v holds M = v + 8*(lane>=16), N = lane&15.
    #pragma unroll
    for (int v = 0; v < 8; ++v) {
        const int m = v + 8 * half;
        lds_h[m * 32 + lm]      = tanhf(c0[v] + b1[lm]);
        lds_h[m * 32 + 16 + lm] = tanhf(c1[v] + b1[16 + lm]);
    }
    __syncthreads();

    // Tiny second GEMM: 16x32 @ 32x6. lane = (row lm, j-group half*3..half*3+2).
    #pragma unroll
    for (int jj = 0; jj < 3; ++jj) {
        const int j = half * 3 + jj;
        float acc = b2[j];
        #pragma unroll
        for (int k = 0; k < 32; ++k)
            acc += lds_h[lm * 32 + k] * w2[k * 6 + j];
        const float p = tanhf(acc);
        float val;
        switch (j) {
            case 0:  val = (p + 1.0f) * -1.5f;            break;  // tmin
            case 1:  val = (p + 1.0f) *  1.5f;            break;  // tmax
            case 2:  val = (p + 1.0f) *  2.5f;            break;  // ddf
            case 3:  val = (p + 1.0f) *  0.05f;           break;  // f = (f+1)/20
            case 4:  val = (p + 1.0f) * 700.0f + 100.0f;  break;  // smax
            default: val = (p + 1.0f) *  20.0f +  10.0f;  break;  // qmax
        }
        params[j * Btot + row0 + lm] = val;
    }
}

// ---------------------------------------------------------------------------
// Kernel B: sequential PRNN scan, one thread per basin, single-wave blocks.
// 4 timesteps per group: 5 aligned b128 loads in, 1 b128 store out.
// Bucket state at the current s1 is carried across steps (the output bucket
// at the new s1 IS the next step's input bucket).
// ---------------------------------------------------------------------------
struct BState {
    float s0, s1;
    float hs1, ov, un, eterm;   // soilbucket terms at current s1
};

__device__ __forceinline__ float
scan_step(BState& st, float p, float tm, float dl,
          float tmin, float tmax, float ddf, float fpar,
          float smax, float qmax, float inv_smax) {
    const float pet = 29.8f * (dl * 24.0f) * 0.611f
                    * __expf(17.3f * tm * __builtin_amdgcn_rcpf(tm + 237.3f))
                    * __builtin_amdgcn_rcpf(tm + 273.2f);

    const float hr    = heav(tm - tmin);
    const float prain = hr * p;
    const float psnow = (1.0f - hr) * p;
    const float dtm   = tm - tmax;
    const float melt  = heav(dtm) * heav(st.s0) * fminf(st.s0, ddf * dtm);

    const float et    = st.hs1 * pet  * (st.ov + st.un * st.s1 * inv_smax);
    const float qsub  = st.hs1 * qmax * (st.ov + st.un * st.eterm);
    const float qsurf = st.hs1 * st.ov * (st.s1 - smax);

    float ds0 = psnow - melt;
    float ds1 = prain + melt - et - qsub - qsurf;
    ds0 = fminf(fmaxf(ds0, -100000.0f), 100000.0f);
    ds1 = fminf(fmaxf(ds1, -100000.0f), 100000.0f);
    st.s0 += ds0;
    st.s1 += ds1;

    // Bucket at the NEW s1: this step's output, next step's input terms.
    const float d = st.s1 - smax;
    st.hs1   = heav(st.s1);
    st.ov    = heav(d);
    st.un    = 1.0f - st.ov;
    st.eterm = __expf(fpar * d);             // exp(-f*(smax-s1))
    return st.hs1 * (qmax * (st.ov + st.un * st.eterm) + st.ov * d);
}

__global__ void __launch_bounds__(32)
prnn_scan_kernel(const float* __restrict__ forcing,
                 const float* __restrict__ params,
                 float* __restrict__ out,
                 int Btot, int T) {
    const int b = blockIdx.x * 32 + threadIdx.x;
    if (b >= Btot) return;

    const float tmin = params[0 * Btot + b];
    const float tmax = params[1 * Btot + b];
    const float ddf  = params[2 * Btot + b];
    const float fpar = params[3 * Btot + b];
    const float smax = params[4 * Btot + b];
    const float qmax = params[5 * Btot + b];
    const float inv_smax = __builtin_amdgcn_rcpf(smax);

    const float* fr = forcing + (size_t)b * (size_t)T * 5u;
    float*       ob = out     + (size_t)b * (size_t)T;
    // b*T*5*4 and b*T*4 are both 16B-aligned (T = 2000).
    const float4* fv  = reinterpret_cast<const float4*>(fr);
    float4*       ov4 = reinterpret_cast<float4*>(ob);

    BState st;
    st.s0 = 0.0f; st.s1 = 0.0f;
    st.hs1   = 0.5f;
    st.ov    = heav(-smax);
    st.un    = 1.0f - st.ov;
    st.eterm = __expf(-fpar * smax);

    const int G = T >> 2;                    // groups of 4 timesteps
    for (int g = 0; g < G; ++g) {
        __builtin_prefetch(fr + 20 * g + 320, 0, 0);   // global_prefetch_b8
        const float4 q0 = fv[5 * g + 0];     // idx  0..3
        const float4 q1 = fv[5 * g + 1];     // idx  4..7
        const float4 q2 = fv[5 * g + 2];     // idx  8..11
        const float4 q3 = fv[5 * g + 3];     // idx 12..15
        const float4 q4 = fv[5 * g + 4];     // idx 16..19

        float4 r;
        r.x = scan_step(st, q0.x, q0.y, q0.z, tmin, tmax, ddf, fpar, smax, qmax, inv_smax);
        r.y = scan_step(st, q1.y, q1.z, q1.w, tmin, tmax, ddf, fpar, smax, qmax, inv_smax);
        r.z = scan_step(st, q2.z, q2.w, q3.x, tmin, tmax, ddf, fpar, smax, qmax, inv_smax);
        r.w = scan_step(st, q3.w, q4.x, q4.y, tmin, tmax, ddf, fpar, smax, qmax, inv_smax);
        ov4[g] = r;
    }
    // Tail (T % 4 != 0) — not taken for T = 2000, kept for generality.
    for (int t = G << 2; t < T; ++t) {
        const float p  = fr[5 * t + 0];
        const float tm = fr[5 * t + 1];
        const float dl = fr[5 * t + 2];
        ob[t] = scan_step(st, p, tm, dl, tmin, tmax, ddf, fpar, smax, qmax, inv_smax);
    }
}

// ---------------------------------------------------------------------------
extern "C" void kernel_launch(void* const* d_in, const int* in_sizes, int n_in,
                              void* d_out, int out_size, void* d_ws, size_t ws_size,
                              hipStream_t stream) {
    const float* forcing = (const float*)d_in[0];
    const float* attrs   = (const float*)d_in[1];
    const float* w1      = (const float*)d_in[2];
    const float* b1      = (const float*)d_in[3];
    const float* w2      = (const float*)d_in[4];
    const float* b2      = (const float*)d_in[5];
    float* out    = (float*)d_out;
    float* params = (float*)d_ws;                 // 6 * Btot floats of scratch

    const int A    = 27;
    const int Btot = in_sizes[1] / A;             // 4096
    const int T    = in_sizes[0] / (Btot * 5);    // 2000

    mlp_params_kernel<<<Btot / 16, 32, 0, stream>>>(attrs, w1, b1, w2, b2,
                                                    params, Btot);
    prnn_scan_kernel<<<(Btot + 31) / 32, 32, 0, stream>>>(forcing, params,
                                                          out, Btot, T);
}